// Net_69913477644625
// MI455X (gfx1250) — compile-verified
//
#include <hip/hip_runtime.h>
#include <math.h>

typedef __attribute__((ext_vector_type(2))) float v2f;
typedef __attribute__((ext_vector_type(8))) float v8f;

// ---------------------------------------------------------------- utilities

__global__ void fill_u32_kernel(unsigned int* __restrict__ p, unsigned int v, long long n) {
  long long stride = (long long)gridDim.x * blockDim.x;
  for (long long t = (long long)blockIdx.x * blockDim.x + threadIdx.x; t < n; t += stride)
    p[t] = v;
}

__device__ inline void atomicMaxFloat(float* addr, float val) {
  // works for mixed-sign floats when addr initialized to -inf bits (0xFF800000)
  if (val >= 0.0f)
    atomicMax((int*)addr, __float_as_int(val));
  else
    atomicMin((unsigned int*)addr, __float_as_uint(val));
}

// ------------------------------------------------- global |d| max over edges

__global__ void edge_absmax_kernel(const float* __restrict__ pos,
                                   const int* __restrict__ row,
                                   const int* __restrict__ col,
                                   const float* __restrict__ ev,
                                   int E, unsigned int* __restrict__ mbits) {
  float lm = 0.0f;
  long long stride = (long long)gridDim.x * blockDim.x;
  for (long long e = (long long)blockIdx.x * blockDim.x + threadIdx.x; e < E; e += stride) {
    float evv = ev ? ev[e] : 1.0f;
    if (evv > 0.0f) {
      int r = row[e], c = col[e];
      float d0 = fabsf(pos[2 * c]     - pos[2 * r]);
      float d1 = fabsf(pos[2 * c + 1] - pos[2 * r + 1]);
      lm = fmaxf(lm, fmaxf(d0, d1));
    }
  }
  __shared__ float red[256];
  red[threadIdx.x] = lm;
  __syncthreads();
  for (int s = blockDim.x >> 1; s > 0; s >>= 1) {
    if ((int)threadIdx.x < s) red[threadIdx.x] = fmaxf(red[threadIdx.x], red[threadIdx.x + s]);
    __syncthreads();
  }
  if (threadIdx.x == 0) atomicMax(mbits, __float_as_uint(red[0]));  // values >= 0
}

// -------------------------------------- B-spline edge scatter (+degree sum)

__global__ void spline_scatter_kernel(const float* __restrict__ x,
                                      const float* __restrict__ pos,
                                      const int* __restrict__ row,
                                      const int* __restrict__ col,
                                      const float* __restrict__ ev,
                                      const unsigned int* __restrict__ mbits,
                                      float* __restrict__ acc,   // [n, 25, Fin]
                                      float* __restrict__ deg,   // [n]
                                      int E, int Fin) {
  long long total = (long long)E * Fin;
  long long stride = (long long)gridDim.x * blockDim.x;
  float m = __uint_as_float(*mbits);
  float inv = 1.0f / (2.0f * m + 1e-12f);
  for (long long t = (long long)blockIdx.x * blockDim.x + threadIdx.x; t < total; t += stride) {
    int e  = (int)(t / Fin);
    int fi = (int)(t % Fin);
    float evv = ev ? ev[e] : 1.0f;
    if (evv <= 0.0f) continue;
    int r = row[e], c = col[e];
    float ps0 = (pos[2 * c]     - pos[2 * r])     * inv + 0.5f;
    float ps1 = (pos[2 * c + 1] - pos[2 * r + 1]) * inv + 0.5f;
    float v0 = ps0 * 4.0f, v1 = ps1 * 4.0f;
    float i0f = fminf(fmaxf(floorf(v0), 0.0f), 3.0f);
    float i1f = fminf(fmaxf(floorf(v1), 0.0f), 3.0f);
    int i0 = (int)i0f, i1 = (int)i1f;
    float f0 = v0 - i0f, f1 = v1 - i1f;
    float xv = x[(size_t)c * Fin + fi];
    float w00 = (1.0f - f0) * (1.0f - f1) * evv;
    float w01 = (1.0f - f0) * f1 * evv;
    float w10 = f0 * (1.0f - f1) * evv;
    float w11 = f0 * f1 * evv;
    size_t base = ((size_t)r * 25) * (size_t)Fin + fi;
    atomicAdd(&acc[base + (size_t)(i0 * 5 + i1)           * Fin], w00 * xv);
    atomicAdd(&acc[base + (size_t)(i0 * 5 + i1 + 1)       * Fin], w01 * xv);
    atomicAdd(&acc[base + (size_t)((i0 + 1) * 5 + i1)     * Fin], w10 * xv);
    atomicAdd(&acc[base + (size_t)((i0 + 1) * 5 + i1 + 1) * Fin], w11 * xv);
    if (fi == 0) atomicAdd(&deg[r], evv);
  }
}

// ---------------------------- dense packing for the WMMA GEMM (branch-free K)

// Apad[m, k] = k < KC        : acc[m,k] / max(deg[m],1)
//             KC <= k < KC+Fin: xin[m, k-KC]
//             else            : 0
__global__ void pack_a_kernel(const float* __restrict__ acc,
                              const float* __restrict__ xin,
                              const float* __restrict__ deg,
                              float* __restrict__ Apad,
                              int n, int Fin, int KC, int Kpad) {
  long long total = (long long)n * Kpad;
  long long stride = (long long)gridDim.x * blockDim.x;
  for (long long t = (long long)blockIdx.x * blockDim.x + threadIdx.x; t < total; t += stride) {
    int m = (int)(t / Kpad);
    int k = (int)(t % Kpad);
    float v;
    if (k < KC)           v = acc[(size_t)m * KC + k] / fmaxf(deg[m], 1.0f);
    else if (k < KC + Fin) v = xin[(size_t)m * Fin + (k - KC)];
    else                  v = 0.0f;
    Apad[t] = v;
  }
}

// Bt[col, k] = k < KC ? W[k,col] : (k < KC+Fin ? root[k-KC,col] : 0)
__global__ void pack_bt_kernel(const float* __restrict__ W,
                               const float* __restrict__ root,
                               float* __restrict__ Bt,
                               int Fout, int Fin, int KC, int Kpad) {
  long long total = (long long)Fout * Kpad;
  long long stride = (long long)gridDim.x * blockDim.x;
  for (long long t = (long long)blockIdx.x * blockDim.x + threadIdx.x; t < total; t += stride) {
    int col = (int)(t / Kpad);
    int k   = (int)(t % Kpad);
    float v;
    if (k < KC)            v = W[(size_t)k * Fout + col];
    else if (k < KC + Fin) v = root[(size_t)(k - KC) * Fout + col];
    else                   v = 0.0f;
    Bt[t] = v;
  }
}

// --------------------- WMMA f32 GEMM: out = ELU(Apad @ Bt^T + bias)
// One wave per 16x16 output tile; inner loop is 2x b64 loads + one
// V_WMMA_F32_16X16X4_F32, unrolled x2.

__global__ void gemm_wmma_kernel(const float* __restrict__ A,   // [n, Kpad]
                                 const float* __restrict__ Bt,  // [Fout, Kpad]
                                 const float* __restrict__ bias,
                                 float* __restrict__ out,       // [n, Fout]
                                 int n, int Fout, int Kpad, int totalTiles) {
  int lane = threadIdx.x & 31;
  int wave = threadIdx.x >> 5;
  int tile = blockIdx.x * (blockDim.x >> 5) + wave;
  if (tile >= totalTiles) return;          // wave-uniform: EXEC stays all-ones
  int ntn   = Fout >> 4;
  int mtile = tile / ntn;
  int ntile = tile - mtile * ntn;
  int lrow  = lane & 15;
  int khalf = (lane >> 4) << 1;            // lanes 16-31 hold K+2, K+3
  const float* arow = A  + (size_t)(mtile * 16 + lrow) * Kpad + khalf;
  const float* brow = Bt + (size_t)(ntile * 16 + lrow) * Kpad + khalf;
  v8f c = {0.f, 0.f, 0.f, 0.f, 0.f, 0.f, 0.f, 0.f};
  int k0 = 0;
  for (; k0 + 8 <= Kpad; k0 += 8) {
    v2f a0 = *(const v2f*)(arow + k0);
    v2f b0 = *(const v2f*)(brow + k0);
    v2f a1 = *(const v2f*)(arow + k0 + 4);
    v2f b1 = *(const v2f*)(brow + k0 + 4);
    c = __builtin_amdgcn_wmma_f32_16x16x4_f32(false, a0, false, b0, (short)0, c,
                                              false, false);
    c = __builtin_amdgcn_wmma_f32_16x16x4_f32(false, a1, false, b1, (short)0, c,
                                              false, false);
  }
  for (; k0 < Kpad; k0 += 4) {
    v2f a = *(const v2f*)(arow + k0);
    v2f b = *(const v2f*)(brow + k0);
    c = __builtin_amdgcn_wmma_f32_16x16x4_f32(false, a, false, b, (short)0, c,
                                              false, false);
  }
  float bc  = bias[ntile * 16 + lrow];
  int   mhi = (lane >> 4) << 3;            // D: VGPR j -> M=j (lo lanes), j+8 (hi)
#pragma unroll
  for (int j = 0; j < 8; ++j) {
    int mm = mtile * 16 + mhi + j;
    float v = c[j] + bc;
    v = v > 0.0f ? v : (__expf(v) - 1.0f); // ELU
    out[(size_t)mm * Fout + (ntile * 16 + lrow)] = v;
  }
}

// ------------------------------------------------------------- grid pooling

__global__ void cluster_kernel(const float* __restrict__ pos,
                               int* __restrict__ cluster,
                               int n, int per_graph, int g, float invsize) {
  long long stride = (long long)gridDim.x * blockDim.x;
  for (long long i = (long long)blockIdx.x * blockDim.x + threadIdx.x; i < n; i += stride) {
    int c0 = (int)floorf(pos[2 * i]     * invsize);
    int c1 = (int)floorf(pos[2 * i + 1] * invsize);
    c0 = min(max(c0, 0), g - 1);
    c1 = min(max(c1, 0), g - 1);
    cluster[i] = ((int)i / per_graph) * (g * g) + c0 * g + c1;
  }
}

__global__ void pool_scatter_kernel(const float* __restrict__ x,
                                    const float* __restrict__ pos,
                                    const float* __restrict__ nv,
                                    const int* __restrict__ cluster,
                                    float* __restrict__ xm,      // [Nc,F] init -inf
                                    float* __restrict__ cnt,     // [Nc]
                                    float* __restrict__ possum,  // [Nc,2]
                                    int n, int F) {
  long long total = (long long)n * F;
  long long stride = (long long)gridDim.x * blockDim.x;
  for (long long t = (long long)blockIdx.x * blockDim.x + threadIdx.x; t < total; t += stride) {
    int i = (int)(t / F);
    int f = (int)(t % F);
    float nvv = nv ? nv[i] : 1.0f;
    if (nvv <= 0.0f) continue;
    int cl = cluster[i];
    atomicMaxFloat(&xm[(size_t)cl * F + f], x[(size_t)i * F + f]);
    if (f == 0) {
      atomicAdd(&cnt[cl], nvv);
      atomicAdd(&possum[2 * cl],     pos[2 * i]     * nvv);
      atomicAdd(&possum[2 * cl + 1], pos[2 * i + 1] * nvv);
    }
  }
}

__global__ void pool_finalize_kernel(const float* __restrict__ xm,
                                     const float* __restrict__ cnt,
                                     const float* __restrict__ possum,
                                     float* __restrict__ x2,
                                     float* __restrict__ pos2,
                                     float* __restrict__ nv2,
                                     int Nc, int F) {
  long long total = (long long)Nc * F;
  long long stride = (long long)gridDim.x * blockDim.x;
  for (long long t = (long long)blockIdx.x * blockDim.x + threadIdx.x; t < total; t += stride) {
    int j = (int)(t / F);
    int f = (int)(t % F);
    float c = cnt[j];
    bool valid = c > 0.0f;
    x2[(size_t)j * F + f] = valid ? xm[(size_t)j * F + f] : 0.0f;
    if (f == 0) {
      nv2[j] = valid ? 1.0f : 0.0f;
      float dc = fmaxf(c, 1.0f);
      pos2[2 * j]     = possum[2 * j]     / dc;
      pos2[2 * j + 1] = possum[2 * j + 1] / dc;
    }
  }
}

// Edge remap + dedup via atomicOr bitmap over (r2,c2) pairs (order-invariant
// equivalent of the reference's sort+drop-duplicates).
__global__ void edge_remap_kernel(const int* __restrict__ row,
                                  const int* __restrict__ col,
                                  const float* __restrict__ ev,
                                  const int* __restrict__ cluster,
                                  unsigned int* __restrict__ bitmap,
                                  long long Nc,
                                  int* __restrict__ row2,
                                  int* __restrict__ col2,
                                  float* __restrict__ ev2, int E) {
  long long stride = (long long)gridDim.x * blockDim.x;
  for (long long e = (long long)blockIdx.x * blockDim.x + threadIdx.x; e < E; e += stride) {
    int r = row[e], c = col[e];
    int r2 = cluster[r], c2 = cluster[c];
    row2[e] = r2;
    col2[e] = c2;
    float evv = ev ? ev[e] : 1.0f;
    float outv = 0.0f;
    if (evv > 0.0f && r2 != c2) {
      long long key = (long long)r2 * Nc + c2;
      unsigned int word = (unsigned int)(key >> 5);
      unsigned int bit  = 1u << (int)(key & 31);
      unsigned int old  = atomicOr(&bitmap[word], bit);
      if (!(old & bit)) outv = 1.0f;  // first edge for this pair is representative
    }
    ev2[e] = outv;
  }
}

// ---------------------------------------- per-graph mean + FC + log_softmax

__global__ void final_head_kernel(const float* __restrict__ h3,   // [4096,128]
                                  const float* __restrict__ nv3,  // [4096]
                                  const float* __restrict__ fcw,  // [128,10]
                                  const float* __restrict__ fcb,  // [10]
                                  float* __restrict__ out) {      // [256,10]
  __shared__ float gfeat[128];
  __shared__ float logits[10];
  int g = blockIdx.x;
  int f = threadIdx.x;  // 0..127
  float s = 0.0f, cntg = 0.0f;
  for (int j = 0; j < 16; ++j) {
    int node = g * 16 + j;
    float nvv = nv3[node];
    s += h3[(size_t)node * 128 + f] * nvv;
    cntg += nvv;
  }
  gfeat[f] = s / fmaxf(cntg, 1.0f);
  __syncthreads();
  if (f < 10) {
    float l = fcb[f];
    for (int k = 0; k < 128; ++k) l += gfeat[k] * fcw[k * 10 + f];
    logits[f] = l;
  }
  __syncthreads();
  if (f == 0) {
    float mx = logits[0];
    for (int o = 1; o < 10; ++o) mx = fmaxf(mx, logits[o]);
    float se = 0.0f;
    for (int o = 0; o < 10; ++o) se += expf(logits[o] - mx);
    float lse = mx + logf(se);
    for (int o = 0; o < 10; ++o) out[g * 10 + o] = logits[o] - lse;
  }
}

// ---------------------------------------------------------------- launcher

static inline long long ll_min(long long a, long long b) { return a < b ? a : b; }

extern "C" void kernel_launch(void* const* d_in, const int* in_sizes, int n_in,
                              void* d_out, int out_size, void* d_ws, size_t ws_size,
                              hipStream_t stream) {
  (void)in_sizes; (void)n_in; (void)out_size; (void)ws_size;
  const float* x     = (const float*)d_in[0];
  const float* posi  = (const float*)d_in[1];
  const int*   eidx  = (const int*)d_in[2];
  const float* W1    = (const float*)d_in[4];
  const float* root1 = (const float*)d_in[5];
  const float* b1    = (const float*)d_in[6];
  const float* W2    = (const float*)d_in[7];
  const float* root2 = (const float*)d_in[8];
  const float* b2    = (const float*)d_in[9];
  const float* W3    = (const float*)d_in[10];
  const float* root3 = (const float*)d_in[11];
  const float* b3    = (const float*)d_in[12];
  const float* fcw   = (const float*)d_in[13];
  const float* fcb   = (const float*)d_in[14];
  float* out = (float*)d_out;

  const int N1 = 65536, E = 1048576;
  const int Nc1 = 16384, Nc2 = 4096;
  const int* row1 = eidx;
  const int* col1 = eidx + E;

  // bump allocator over workspace
  char* ws = (char*)d_ws;
  size_t off = 0;
  auto alloc = [&](size_t bytes) -> void* {
    void* p = ws + off;
    off = (off + bytes + 255) & ~(size_t)255;
    return p;
  };
  unsigned int* mbits = (unsigned int*)alloc(3 * sizeof(unsigned int));
  float* deg1  = (float*)alloc((size_t)N1 * 4);
  float* acc1  = (float*)alloc((size_t)N1 * 25 * 4);
  float* h1    = (float*)alloc((size_t)N1 * 32 * 4);
  int*   clu1  = (int*)alloc((size_t)N1 * 4);
  float* cnt1  = (float*)alloc((size_t)Nc1 * 4);
  float* psum1 = (float*)alloc((size_t)Nc1 * 2 * 4);
  float* xm1   = (float*)alloc((size_t)Nc1 * 32 * 4);
  float* h1p   = (float*)alloc((size_t)Nc1 * 32 * 4);
  float* pos2  = (float*)alloc((size_t)Nc1 * 2 * 4);
  float* nv2   = (float*)alloc((size_t)Nc1 * 4);
  int*   row2  = (int*)alloc((size_t)E * 4);
  int*   col2  = (int*)alloc((size_t)E * 4);
  float* ev2   = (float*)alloc((size_t)E * 4);
  float* deg2  = (float*)alloc((size_t)Nc1 * 4);
  float* h2    = (float*)alloc((size_t)Nc1 * 64 * 4);
  int*   clu2  = (int*)alloc((size_t)Nc1 * 4);
  float* cnt2  = (float*)alloc((size_t)Nc2 * 4);
  float* psum2 = (float*)alloc((size_t)Nc2 * 2 * 4);
  float* xm2   = (float*)alloc((size_t)Nc2 * 64 * 4);
  float* h2p   = (float*)alloc((size_t)Nc2 * 64 * 4);
  float* pos3  = (float*)alloc((size_t)Nc2 * 2 * 4);
  float* nv3   = (float*)alloc((size_t)Nc2 * 4);
  int*   row3  = (int*)alloc((size_t)E * 4);
  int*   col3  = (int*)alloc((size_t)E * 4);
  float* ev3   = (float*)alloc((size_t)E * 4);
  float* deg3  = (float*)alloc((size_t)Nc2 * 4);
  float* h3    = (float*)alloc((size_t)Nc2 * 128 * 4);
  // BIG region: time-shared bitmap1(32MB) -> acc2(52.4MB) -> bitmap2(2MB) -> acc3(26.2MB)
  void*  BIG   = alloc((size_t)Nc1 * 800 * 4);
  // packed GEMM operands (reused per layer)
  float* APAD  = (float*)alloc((size_t)Nc1 * 832 * 4);   // >= all layers' [n,Kpad]
  float* BT    = (float*)alloc((size_t)128 * 1664 * 4);  // >= all layers' [Fout,Kpad]

  auto fill = [&](void* p, unsigned int v, long long n) {
    int blocks = (int)ll_min((n + 255) / 256, 65536);
    fill_u32_kernel<<<blocks, 256, 0, stream>>>((unsigned int*)p, v, n);
  };
  auto gridFor = [&](long long n) { return (int)ll_min((n + 255) / 256, 65536); };

  // ---------------- Layer 1 (N1 nodes, Fin=1 -> Fout=32, Kpad=28) ----------------
  fill(mbits, 0u, 3);
  fill(deg1, 0u, N1);
  fill(acc1, 0u, (long long)N1 * 25);
  edge_absmax_kernel<<<gridFor(E), 256, 0, stream>>>(posi, row1, col1, nullptr, E, mbits + 0);
  spline_scatter_kernel<<<gridFor((long long)E * 1), 256, 0, stream>>>(
      x, posi, row1, col1, nullptr, mbits + 0, acc1, deg1, E, 1);
  pack_a_kernel<<<gridFor((long long)N1 * 28), 256, 0, stream>>>(acc1, x, deg1, APAD, N1, 1, 25, 28);
  pack_bt_kernel<<<gridFor((long long)32 * 28), 256, 0, stream>>>(W1, root1, BT, 32, 1, 25, 28);
  {
    int tiles = (N1 / 16) * (32 / 16);
    gemm_wmma_kernel<<<(tiles + 3) / 4, 128, 0, stream>>>(APAD, BT, b1, h1, N1, 32, 28, tiles);
  }

  // ---------------- Pool 1 (size 4 -> g=8, Nc1=16384) ----------------
  cluster_kernel<<<gridFor(N1), 256, 0, stream>>>(posi, clu1, N1, 256, 8, 0.25f);
  fill(xm1, 0xFF800000u, (long long)Nc1 * 32);
  fill(cnt1, 0u, Nc1);
  fill(psum1, 0u, (long long)Nc1 * 2);
  pool_scatter_kernel<<<gridFor((long long)N1 * 32), 256, 0, stream>>>(
      h1, posi, nullptr, clu1, xm1, cnt1, psum1, N1, 32);
  pool_finalize_kernel<<<gridFor((long long)Nc1 * 32), 256, 0, stream>>>(
      xm1, cnt1, psum1, h1p, pos2, nv2, Nc1, 32);
  fill(BIG, 0u, (long long)Nc1 * Nc1 / 32);  // bitmap1
  edge_remap_kernel<<<gridFor(E), 256, 0, stream>>>(
      row1, col1, nullptr, clu1, (unsigned int*)BIG, Nc1, row2, col2, ev2, E);

  // ---------------- Layer 2 (Nc1 nodes, 32 -> 64, Kpad=832) ----------------
  edge_absmax_kernel<<<gridFor(E), 256, 0, stream>>>(pos2, row2, col2, ev2, E, mbits + 1);
  fill(deg2, 0u, Nc1);
  fill(BIG, 0u, (long long)Nc1 * 800);  // acc2
  spline_scatter_kernel<<<gridFor((long long)E * 32), 256, 0, stream>>>(
      h1p, pos2, row2, col2, ev2, mbits + 1, (float*)BIG, deg2, E, 32);
  pack_a_kernel<<<gridFor((long long)Nc1 * 832), 256, 0, stream>>>(
      (float*)BIG, h1p, deg2, APAD, Nc1, 32, 800, 832);
  pack_bt_kernel<<<gridFor((long long)64 * 832), 256, 0, stream>>>(W2, root2, BT, 64, 32, 800, 832);
  {
    int tiles = (Nc1 / 16) * (64 / 16);
    gemm_wmma_kernel<<<(tiles + 3) / 4, 128, 0, stream>>>(APAD, BT, b2, h2, Nc1, 64, 832, tiles);
  }

  // ---------------- Pool 2 (size 8 -> g=4, Nc2=4096) ----------------
  cluster_kernel<<<gridFor(Nc1), 256, 0, stream>>>(pos2, clu2, Nc1, 64, 4, 0.125f);
  fill(xm2, 0xFF800000u, (long long)Nc2 * 64);
  fill(cnt2, 0u, Nc2);
  fill(psum2, 0u, (long long)Nc2 * 2);
  pool_scatter_kernel<<<gridFor((long long)Nc1 * 64), 256, 0, stream>>>(
      h2, pos2, nv2, clu2, xm2, cnt2, psum2, Nc1, 64);
  pool_finalize_kernel<<<gridFor((long long)Nc2 * 64), 256, 0, stream>>>(
      xm2, cnt2, psum2, h2p, pos3, nv3, Nc2, 64);
  fill(BIG, 0u, (long long)Nc2 * Nc2 / 32);  // bitmap2
  edge_remap_kernel<<<gridFor(E), 256, 0, stream>>>(
      row2, col2, ev2, clu2, (unsigned int*)BIG, Nc2, row3, col3, ev3, E);

  // ---------------- Layer 3 (Nc2 nodes, 64 -> 128, Kpad=1664) ----------------
  edge_absmax_kernel<<<gridFor(E), 256, 0, stream>>>(pos3, row3, col3, ev3, E, mbits + 2);
  fill(deg3, 0u, Nc2);
  fill(BIG, 0u, (long long)Nc2 * 1600);  // acc3
  spline_scatter_kernel<<<gridFor((long long)E * 64), 256, 0, stream>>>(
      h2p, pos3, row3, col3, ev3, mbits + 2, (float*)BIG, deg3, E, 64);
  pack_a_kernel<<<gridFor((long long)Nc2 * 1664), 256, 0, stream>>>(
      (float*)BIG, h2p, deg3, APAD, Nc2, 64, 1600, 1664);
  pack_bt_kernel<<<gridFor((long long)128 * 1664), 256, 0, stream>>>(W3, root3, BT, 128, 64, 1600, 1664);
  {
    int tiles = (Nc2 / 16) * (128 / 16);
    gemm_wmma_kernel<<<(tiles + 3) / 4, 128, 0, stream>>>(APAD, BT, b3, h3, Nc2, 128, 1664, tiles);
  }

  // ---------------- Global mean pool + FC + log_softmax ----------------
  final_head_kernel<<<256, 128, 0, stream>>>(h3, nv3, fcw, fcb, out);
}